// BiLSTMCRF_82841329205414
// MI455X (gfx1250) — compile-verified
//
#include <hip/hip_runtime.h>
#include <hip/hip_bf16.h>
#include <math.h>

// ---------------------------------------------------------------------------
// BiLSTM-CRF forward for gfx1250 (MI455X), fp32 WMMA (v_wmma_f32_16x16x4_f32).
//
// Shapes: SEQ=256, B=32, WL=16, CE=64, ch=64 (CH=128), WE=200, wh=128 (WH=256), T=9
// Key trick: cfeat = concat(yf[15], yb[0])  => backward char LSTM = 1 step only.
//
// Round-1: LICM hoisted loop-invariant WMMA B-fragments -> scratch spills.
// Round-2: opaque pointer fixed spills, but killed addrspace inference ->
//          flat_load_b64 (uses LOADcnt *and* DScnt, coupling weight streams
//          with the LDS A-fragment waits). Now: keep the global pointer intact
//          and add an opaque *zero offset* per iteration instead -> addresses
//          stay non-invariant (no LICM/spill) but loads select global_load_b64
//          (LOADcnt only).
// ---------------------------------------------------------------------------

typedef __attribute__((ext_vector_type(2))) float v2f;
typedef __attribute__((ext_vector_type(8))) float v8f;

__device__ __forceinline__ v8f wmma4(v2f a, v2f b, v8f c) {
  // D = A(16x4) * B(4x16) + C(16x16), fp32, wave32.
  return __builtin_amdgcn_wmma_f32_16x16x4_f32(
      /*neg_a=*/false, a, /*neg_b=*/false, b,
      /*c_mod=*/(short)0, c, /*reuse_a=*/false, /*reuse_b=*/false);
}

__device__ __forceinline__ float sigf(float x) { return 1.0f / (1.0f + expf(-x)); }

// Opaque zero (SGPR): re-materialized every call site; adding it to an index
// makes addresses non-provably-loop-invariant (defeats LICM -> no spills)
// while keeping pointer provenance (global addrspace -> global_load_b64).
__device__ __forceinline__ size_t opaque_zero() {
  size_t z = 0;
  asm volatile("" : "+s"(z));
  return z;
}

// ---------------- workspace layout (float offsets) -------------------------
#define WS_CFEAT   ((size_t)0)            // 8192*128      = 1,048,576
#define WS_GX      ((size_t)1048576)      // 2*8192*512    = 8,388,608
#define WS_WOUT    ((size_t)9437184)      // 8192*256      = 2,097,152
#define WS_EM      ((size_t)11534336)     // 8192*9        = 73,728
#define WS_BPC     ((size_t)11608064)     // 2*512*64      = 65,536
#define WS_BPIH    ((size_t)11673600)     // 2*2624*64     = 335,872
#define WS_BPHH    ((size_t)12009472)     // 2*1024*64     = 131,072
#define WS_HIST    ((size_t)12140544)     // 255*32*9 ints = 73,440
// total ~48.9 MB

// ---------------------------------------------------------------------------
// Pack (N,K1+K2) row-major weight(s) into WMMA B fragments:
// frag f = nt*ksteps+ks ; out[f*64 + lane*2 + e] = W[n][k+e],
// n = nt*16 + (lane&15), k = ks*4 + (lane>>4)*2.  (K2==0 => single matrix)
// ---------------------------------------------------------------------------
__global__ void pack_w2_kernel(const float* __restrict__ W1, const float* __restrict__ W2,
                               float* __restrict__ out, int N, int K1, int K2) {
  int ksteps = (K1 + K2) >> 2;
  int total  = (N >> 4) * ksteps;
  int frag = blockIdx.x * (blockDim.x >> 5) + (threadIdx.x >> 5);
  int lane = threadIdx.x & 31;
  if (frag >= total) return;
  int nt = frag / ksteps, ks = frag - nt * ksteps;
  int n = nt * 16 + (lane & 15);
  int k = ks * 4 + (lane >> 4) * 2;
#pragma unroll
  for (int e = 0; e < 2; ++e) {
    int kk = k + e;
    float v = (kk < K1) ? W1[(size_t)n * K1 + kk] : W2[(size_t)n * K2 + (kk - K1)];
    out[(size_t)frag * 64 + lane * 2 + e] = v;
  }
}

// ---------------------------------------------------------------------------
// Char BiLSTM. grid(512, 2): x = 16-word tile, y = direction. block = 128 (4 waves).
// Fused gate GEMM: A = [emb_t(64) | h(64)] (16 x 128), B = [Wih|Whh] (K=128, N=256).
// Wave w owns gate tiles nt = {w, w+4, w+8, w+12} => i/f/g/o of units j=w*16+(lane&15)
// all in-lane; cell state c lives in 8 VGPRs per lane for the whole sequence.
// Backward direction runs exactly one step on x[15] (yields yb[0]).
// ---------------------------------------------------------------------------
__global__ void __launch_bounds__(128, 1)
char_lstm_kernel(const int* __restrict__ chars,      // (8192,16)
                 const float* __restrict__ char_emb, // (100,64)
                 const float* __restrict__ Bpack,    // [2][512*64]
                 const float* __restrict__ b_f, const float* __restrict__ b_b,
                 const float* __restrict__ h0, const float* __restrict__ c0, // (2,8192,64)
                 float* __restrict__ cfeat) {        // (8192,128)
  __shared__ float aT[16][128];  // [:,0:64]=emb_t, [:,64:128]=h
  int dir = blockIdx.y;
  int wbase = blockIdx.x * 16;
  int tid = threadIdx.x, wave = tid >> 5, lane = tid & 31;
  int half = lane >> 4, mr = lane & 15;
  int j = wave * 16 + mr;  // hidden unit 0..63
  const float* bias = dir ? b_b : b_f;
  const float* Bp = Bpack + (size_t)dir * (16 * 32 * 64);

  float creg[8];
#pragma unroll
  for (int r = 0; r < 8; ++r)
    creg[r] = c0[(size_t)(dir * 8192 + wbase + half * 8 + r) * 64 + j];
  float bv[4];
#pragma unroll
  for (int g = 0; g < 4; ++g) bv[g] = bias[(wave + g * 4) * 16 + mr];
  for (int i = tid; i < 16 * 64; i += 128) {
    int m = i >> 6, jj = i & 63;
    aT[m][64 + jj] = h0[(size_t)(dir * 8192 + wbase + m) * 64 + jj];
  }

  int nsteps = dir ? 1 : 16;
  for (int s = 0; s < nsteps; ++s) {
    int t = dir ? 15 : s;
    __syncthreads();
    for (int i = tid; i < 16 * 64; i += 128) {
      int m = i >> 6, jj = i & 63;
      int ci = chars[(size_t)(wbase + m) * 16 + t];
      aT[m][jj] = char_emb[(size_t)ci * 64 + jj];
    }
    __syncthreads();
    const float* BpI = Bp + opaque_zero();  // in-loop weight streaming, global_load
    v8f acc[4];
#pragma unroll
    for (int g = 0; g < 4; ++g) {
      int nt = wave + g * 4;          // i:0-3 f:4-7 g:8-11 o:12-15
      v8f c;
#pragma unroll
      for (int r = 0; r < 8; ++r) c[r] = bv[g];
      for (int ks = 0; ks < 32; ++ks) {
        v2f a = *(const v2f*)&aT[mr][ks * 4 + half * 2];
        v2f b = *(const v2f*)&BpI[(size_t)(nt * 32 + ks) * 64 + lane * 2];
        c = wmma4(a, b, c);
      }
      acc[g] = c;
    }
    __syncthreads();
#pragma unroll
    for (int r = 0; r < 8; ++r) {
      int m = half * 8 + r;
      float ig = acc[0][r], fg = acc[1][r], gg = acc[2][r], og = acc[3][r];
      float cc = sigf(fg) * creg[r] + sigf(ig) * tanhf(gg);
      float hh = sigf(og) * tanhf(cc);
      creg[r] = cc;
      aT[m][64 + j] = hh;
    }
  }
  __syncthreads();
  for (int i = tid; i < 16 * 64; i += 128) {
    int m = i >> 6, jj = i & 63;
    cfeat[(size_t)(wbase + m) * 128 + dir * 64 + jj] = aT[m][64 + jj];
  }
}

// ---------------------------------------------------------------------------
// Word input GEMM (hoisted): Gx[dir][t*32+b][n] = x[t,b]·Wih_dirᵀ + b_dir.
// x row = [word_emb[sent] (200) | cfeat (128)], K=328 (82 k-steps), N=512.
// grid(512, 2) — 16-row M tiles; block 256 (8 waves × 4 N-tiles).
// ---------------------------------------------------------------------------
__global__ void __launch_bounds__(256, 1)
word_input_gemm_kernel(const int* __restrict__ sent,       // (8192)
                       const float* __restrict__ word_emb, // (50000,200)
                       const float* __restrict__ cfeat,    // (8192,128)
                       const float* __restrict__ Bpack_ih, // [2][2624*64]
                       const float* __restrict__ b_f, const float* __restrict__ b_b,
                       float* __restrict__ Gx) {           // [2][8192][512]
  __shared__ float aT[16][328];
  int dir = blockIdx.y;
  int rbase = blockIdx.x * 16;
  int tid = threadIdx.x, wave = tid >> 5, lane = tid & 31;
  int half = lane >> 4, mr = lane & 15;

  for (int i = tid; i < 16 * 328; i += 256) {
    int m = i / 328, k = i - m * 328;
    int row = rbase + m;
    float v = (k < 200) ? word_emb[(size_t)sent[row] * 200 + k]
                        : cfeat[(size_t)row * 128 + (k - 200)];
    aT[m][k] = v;
  }
  __syncthreads();

  const float* Bp = Bpack_ih + (size_t)dir * (32 * 82 * 64);
  const float* bias = dir ? b_b : b_f;
#pragma unroll 1
  for (int tt = 0; tt < 4; ++tt) {
    int tn = wave * 4 + tt;
    int n = tn * 16 + mr;
    float bvv = bias[n];
    v8f c;
#pragma unroll
    for (int r = 0; r < 8; ++r) c[r] = bvv;
    const float* BpI = Bp + opaque_zero();
    for (int ks = 0; ks < 82; ++ks) {
      v2f a = *(const v2f*)&aT[mr][ks * 4 + half * 2];
      v2f b = *(const v2f*)&BpI[((size_t)tn * 82 + ks) * 64 + lane * 2];
      c = wmma4(a, b, c);
    }
#pragma unroll
    for (int r = 0; r < 8; ++r) {
      int m = half * 8 + r;
      Gx[((size_t)dir * 8192 + rbase + m) * 512 + n] = c[r];
    }
  }
}

// ---------------------------------------------------------------------------
// Word recurrence: 256 sequential steps per direction. grid(2) = directions,
// block 512 (16 waves). gates = Gx[t] + h·Whhᵀ ; M=32 (2 tiles), N=512, K=128.
// Wave w: mt = w>>3, owns nt = {(w&7), +8, +16, +24} => in-lane i/f/g/o for
// unit j = (w&7)*16 + (lane&15); c state stays in registers, h in LDS.
// ---------------------------------------------------------------------------
__global__ void __launch_bounds__(512, 1)
word_recurrent_kernel(const float* __restrict__ Gx,       // [2][8192][512]
                      const float* __restrict__ Bpack_hh, // [2][1024*64]
                      const float* __restrict__ h0, const float* __restrict__ c0, // (2,32,128)
                      float* __restrict__ wout) {          // (8192,256)
  __shared__ float hT[32][128];
  int dir = blockIdx.x;
  int tid = threadIdx.x, wave = tid >> 5, lane = tid & 31;
  int half = lane >> 4, mr = lane & 15;
  int mt = wave >> 3;
  int w2 = wave & 7;
  int j = w2 * 16 + mr;  // hidden unit 0..127

  float creg[8];
#pragma unroll
  for (int r = 0; r < 8; ++r)
    creg[r] = c0[(size_t)(dir * 32 + mt * 16 + half * 8 + r) * 128 + j];
  for (int i = tid; i < 32 * 128; i += 512) {
    int m = i >> 7, jj = i & 127;
    hT[m][jj] = h0[(size_t)(dir * 32 + m) * 128 + jj];
  }

  const float* Bp = Bpack_hh + (size_t)dir * (32 * 32 * 64);
  const float* GxD = Gx + (size_t)dir * 8192 * 512;

  for (int s = 0; s < 256; ++s) {
    int t = dir ? (255 - s) : s;
    __syncthreads();  // h writes from previous step visible
    const float* BpI = Bp + opaque_zero();  // Whh streams from WGP$/L2 each step
    v8f acc[4];
#pragma unroll
    for (int g = 0; g < 4; ++g) {
      int nt = w2 + g * 8;            // i:0-7 f:8-15 g:16-23 o:24-31
      int n = nt * 16 + mr;
      v8f c;
#pragma unroll
      for (int r = 0; r < 8; ++r)
        c[r] = GxD[((size_t)t * 32 + mt * 16 + half * 8 + r) * 512 + n];
      for (int ks = 0; ks < 32; ++ks) {
        v2f a = *(const v2f*)&hT[mt * 16 + mr][ks * 4 + half * 2];
        v2f b = *(const v2f*)&BpI[(size_t)(nt * 32 + ks) * 64 + lane * 2];
        c = wmma4(a, b, c);
      }
      acc[g] = c;
    }
    __syncthreads();  // all hT reads done before overwrite
#pragma unroll
    for (int r = 0; r < 8; ++r) {
      int m = mt * 16 + half * 8 + r;
      float ig = acc[0][r], fg = acc[1][r], gg = acc[2][r], og = acc[3][r];
      float cc = sigf(fg) * creg[r] + sigf(ig) * tanhf(gg);
      float hh = sigf(og) * tanhf(cc);
      creg[r] = cc;
      hT[m][j] = hh;
      wout[((size_t)t * 32 + m) * 256 + dir * 128 + j] = hh;
    }
  }
}

// ---------------------------------------------------------------------------
// Emissions: em[row][tag] = wout[row]·out_W[tag] + out_b[tag] ; 8192x9, K=256.
// ---------------------------------------------------------------------------
__global__ void emissions_kernel(const float* __restrict__ wout, const float* __restrict__ W,
                                 const float* __restrict__ b, float* __restrict__ em) {
  int idx = blockIdx.x * blockDim.x + threadIdx.x;
  if (idx >= 8192 * 9) return;
  int row = idx / 9, tag = idx - row * 9;
  const float* wr = wout + (size_t)row * 256;
  const float* wt = W + (size_t)tag * 256;
  float s = b[tag];
  for (int k = 0; k < 256; ++k) s = fmaf(wr[k], wt[k], s);
  em[idx] = s;
}

// ---------------------------------------------------------------------------
// CRF log-likelihood (masked forward alg) + Viterbi. One block, 288 threads:
// thread = (batch b = tid/9, tag j = tid%9). out[0]=score, out[1+t*32+b]=path.
// ---------------------------------------------------------------------------
__global__ void crf_viterbi_kernel(const float* __restrict__ em,   // (256,32,9)
                                   const int* __restrict__ tags,   // (256,32)
                                   const int* __restrict__ mask,   // (256,32)
                                   const float* __restrict__ start,
                                   const float* __restrict__ endv,
                                   const float* __restrict__ trans, // (9,9)
                                   int* __restrict__ hist,          // (255,32,9)
                                   float* __restrict__ out) {
  __shared__ float alpha[32][9];
  __shared__ float vs[32][9];
  __shared__ float tr[81];
  __shared__ float st[9], en[9];
  __shared__ float red[32];
  __shared__ int bestl[32];
  int tid = threadIdx.x;
  if (tid < 81) tr[tid] = trans[tid];
  if (tid < 9) { st[tid] = start[tid]; en[tid] = endv[tid]; }
  __syncthreads();
  int b = tid / 9, j = tid - b * 9;

  float a0 = st[j] + em[b * 9 + j];
  alpha[b][j] = a0;
  vs[b][j] = a0;
  __syncthreads();

  for (int t = 1; t < 256; ++t) {
    float e = em[((size_t)t * 32 + b) * 9 + j];
    int m = mask[t * 32 + b];
    float mx = -3.0e38f;
#pragma unroll
    for (int i = 0; i < 9; ++i) { float v = alpha[b][i] + tr[i * 9 + j]; mx = v > mx ? v : mx; }
    float ssum = 0.f;
#pragma unroll
    for (int i = 0; i < 9; ++i) ssum += expf(alpha[b][i] + tr[i * 9 + j] - mx);
    float na = mx + logf(ssum) + e;
    float bm = -3.0e38f; int arg = 0;
#pragma unroll
    for (int i = 0; i < 9; ++i) { float v = vs[b][i] + tr[i * 9 + j]; if (v > bm) { bm = v; arg = i; } }
    float nv = bm + e;
    __syncthreads();
    if (m) { alpha[b][j] = na; vs[b][j] = nv; }
    hist[((size_t)(t - 1) * 32 + b) * 9 + j] = arg;
    __syncthreads();
  }

  if (j == 0) {
    float bm = -3.0e38f; int bi = 0;
#pragma unroll
    for (int i = 0; i < 9; ++i) { float v = vs[b][i] + en[i]; if (v > bm) { bm = v; bi = i; } }
    bestl[b] = bi;
  }
  __syncthreads();

  if (tid < 32) {
    int bb = tid;
    // Viterbi backtrack
    int bst = bestl[bb];
    out[1 + 255 * 32 + bb] = (float)bst;
    for (int t = 254; t >= 0; --t) {
      int prev = hist[((size_t)t * 32 + bb) * 9 + bst];
      if (mask[(t + 1) * 32 + bb]) bst = prev;
      out[1 + t * 32 + bb] = (float)bst;
    }
    // numerator
    int tg = tags[bb];
    float num = st[tg] + em[bb * 9 + tg];
    int prevtag = tg;
    int cnt = mask[bb] ? 1 : 0;
    for (int t = 1; t < 256; ++t) {
      int tgt = tags[t * 32 + bb];
      if (mask[t * 32 + bb]) { num += em[((size_t)t * 32 + bb) * 9 + tgt] + tr[prevtag * 9 + tgt]; cnt++; }
      prevtag = tgt;
    }
    num += en[tags[(cnt - 1) * 32 + bb]];
    // logZ
    float mx = -3.0e38f;
    for (int i = 0; i < 9; ++i) { float v = alpha[bb][i] + en[i]; mx = v > mx ? v : mx; }
    float ssum = 0.f;
    for (int i = 0; i < 9; ++i) ssum += expf(alpha[bb][i] + en[i] - mx);
    red[bb] = num - (mx + logf(ssum));
  }
  __syncthreads();
  if (tid == 0) {
    float s = 0.f;
    for (int bb = 0; bb < 32; ++bb) s += red[bb];
    out[0] = s;
  }
}

// ---------------------------------------------------------------------------
extern "C" void kernel_launch(void* const* d_in, const int* in_sizes, int n_in,
                              void* d_out, int out_size, void* d_ws, size_t ws_size,
                              hipStream_t stream) {
  (void)in_sizes; (void)n_in; (void)out_size; (void)ws_size;
  // setup_inputs() dict order, params flattened recursively in insertion order
  const int*   sent   = (const int*)d_in[0];     // (256,32)
  const int*   chars  = (const int*)d_in[1];     // (256,32,16)
  const int*   tags   = (const int*)d_in[2];     // (256,32)
  const int*   mask   = (const int*)d_in[3];     // (256,32)
  const float* cemb   = (const float*)d_in[4];   // (100,64)
  const float* wemb   = (const float*)d_in[5];   // (50000,200)
  const float* cWih_f = (const float*)d_in[6];   // (256,64)
  const float* cWhh_f = (const float*)d_in[7];   // (256,64)
  const float* cb_f   = (const float*)d_in[8];   // (256)
  const float* cWih_b = (const float*)d_in[9];
  const float* cWhh_b = (const float*)d_in[10];
  const float* cb_b   = (const float*)d_in[11];
  const float* wWih_f = (const float*)d_in[12];  // (512,328)
  const float* wWhh_f = (const float*)d_in[13];  // (512,128)
  const float* wb_f   = (const float*)d_in[14];  // (512)
  const float* wWih_b = (const float*)d_in[15];
  const float* wWhh_b = (const float*)d_in[16];
  const float* wb_b   = (const float*)d_in[17];
  const float* outW   = (const float*)d_in[18];  // (9,256)
  const float* outb   = (const float*)d_in[19];  // (9)
  const float* crf_s  = (const float*)d_in[20];  // (9)
  const float* crf_e  = (const float*)d_in[21];  // (9)
  const float* crf_t  = (const float*)d_in[22];  // (9,9)
  const float* ch0    = (const float*)d_in[23];  // (2,8192,64)
  const float* cc0    = (const float*)d_in[24];
  const float* wh0    = (const float*)d_in[25];  // (2,32,128)
  const float* wc0    = (const float*)d_in[26];

  float* ws    = (float*)d_ws;
  float* cfeat = ws + WS_CFEAT;
  float* gx    = ws + WS_GX;
  float* wout  = ws + WS_WOUT;
  float* emw   = ws + WS_EM;
  float* bpc   = ws + WS_BPC;
  float* bpih  = ws + WS_BPIH;
  float* bphh  = ws + WS_BPHH;
  int*   hist  = (int*)(ws + WS_HIST);
  float* out   = (float*)d_out;

  // 1) pack weights into WMMA B fragments (char: fused [Wih|Whh], K=64+64)
  pack_w2_kernel<<<64, 256, 0, stream>>>(cWih_f, cWhh_f, bpc,           256, 64, 64);
  pack_w2_kernel<<<64, 256, 0, stream>>>(cWih_b, cWhh_b, bpc + 32768,   256, 64, 64);
  pack_w2_kernel<<<328, 256, 0, stream>>>(wWih_f, wWih_f, bpih,          512, 328, 0);
  pack_w2_kernel<<<328, 256, 0, stream>>>(wWih_b, wWih_b, bpih + 167936, 512, 328, 0);
  pack_w2_kernel<<<128, 256, 0, stream>>>(wWhh_f, wWhh_f, bphh,          512, 128, 0);
  pack_w2_kernel<<<128, 256, 0, stream>>>(wWhh_b, wWhh_b, bphh + 65536,  512, 128, 0);

  // 2) char BiLSTM -> cfeat (fwd 16 steps, bwd 1 step)
  char_lstm_kernel<<<dim3(512, 2), 128, 0, stream>>>(chars, cemb, bpc, cb_f, cb_b,
                                                     ch0, cc0, cfeat);
  // 3) hoisted word-input GEMM -> Gx
  word_input_gemm_kernel<<<dim3(512, 2), 256, 0, stream>>>(sent, wemb, cfeat, bpih,
                                                           wb_f, wb_b, gx);
  // 4) word recurrence (persistent, 1 WG per direction) -> wout
  word_recurrent_kernel<<<2, 512, 0, stream>>>(gx, bphh, wh0, wc0, wout);
  // 5) emissions -> em
  emissions_kernel<<<288, 256, 0, stream>>>(wout, outW, outb, emw);
  // 6) CRF score + Viterbi path -> d_out
  crf_viterbi_kernel<<<1, 288, 0, stream>>>(emw, tags, mask, crf_s, crf_e, crf_t,
                                            hist, out);
}